// SymmetricAttentionSOTW_27187142984173
// MI455X (gfx1250) — compile-verified
//
#include <hip/hip_runtime.h>
#include <math.h>

// ---------------------------------------------------------------------------
// Problem constants (from reference): B=32, L=128, D=512
// Flattened GEMM shapes: M = B*L = 4096 rows, K = 512, N = 512.
// ---------------------------------------------------------------------------
#define MDIM 4096
#define NDIM 512
#define KDIM 512
#define LSEQ 128

typedef float v2f __attribute__((ext_vector_type(2)));
typedef float v8f __attribute__((ext_vector_type(8)));

// ---------------------------------------------------------------------------
// Hardware tanh: 4 independent v_tanh_f32 on the TRANS pipe. Each trans op is
// followed by an independent trans op (satisfies the 1-independent-op rule,
// ISA 7.4); trailing v_nop covers the last result before the consumer FMA.
// ---------------------------------------------------------------------------
__device__ __forceinline__ float4 htanh4(float4 x) {
    float4 r;
    asm volatile(
        "v_tanh_f32 %0, %4\n\t"
        "v_tanh_f32 %1, %5\n\t"
        "v_tanh_f32 %2, %6\n\t"
        "v_tanh_f32 %3, %7\n\t"
        "v_nop"
        : "=&v"(r.x), "=&v"(r.y), "=&v"(r.z), "=&v"(r.w)
        : "v"(x.x), "v"(x.y), "v"(x.z), "v"(x.w));
    return r;
}

// ---------------------------------------------------------------------------
// FP32 WMMA GEMM: Out[M,N] = A[M,K] @ W[K,N] (+ bias[n]) (+= Out if accum)
// One wave computes a 16x64 strip via 4x V_WMMA_F32_16X16X4_F32 per K-step.
// 4 waves / block -> 64x64 tile. Grid (M/64, N/64) = (64, 8).
// A-frag (16x4 f32): lanes 0-15 M=lane K={0,1}; lanes 16-31 M=lane-16 K={2,3}.
// B-frag (4x16 f32): mirrored (K split across lane halves, N across lanes).
// C/D: vgpr r -> row m0 + r + 8*(lane>=16), col n0 + (lane&15).
// ---------------------------------------------------------------------------
__global__ __launch_bounds__(128)
void sattn_gemm_wmma_f32(const float* __restrict__ A,
                         const float* __restrict__ W,
                         const float* __restrict__ bias,
                         float* __restrict__ Out,
                         int accum)
{
    const int wave    = threadIdx.x >> 5;   // 0..3
    const int lane    = threadIdx.x & 31;
    const int l16     = lane & 15;
    const int halfsel = lane >> 4;          // 0 or 1

    const int m0 = blockIdx.x * 64 + wave * 16;
    const int n0 = blockIdx.y * 64;

    v8f acc0 = {}; v8f acc1 = {}; v8f acc2 = {}; v8f acc3 = {};

    const float* Arow = A + (size_t)(m0 + l16) * KDIM + 2 * halfsel;

    for (int k = 0; k < KDIM; k += 4) {
        v2f a;
        a.x = Arow[k + 0];
        a.y = Arow[k + 1];

        const float* Wp = W + (size_t)(k + 2 * halfsel) * NDIM + n0 + l16;

        v2f b0; b0.x = Wp[0];  b0.y = Wp[NDIM];
        v2f b1; b1.x = Wp[16]; b1.y = Wp[NDIM + 16];
        v2f b2; b2.x = Wp[32]; b2.y = Wp[NDIM + 32];
        v2f b3; b3.x = Wp[48]; b3.y = Wp[NDIM + 48];

        acc0 = __builtin_amdgcn_wmma_f32_16x16x4_f32(false, a, false, b0, (short)0, acc0, false, false);
        acc1 = __builtin_amdgcn_wmma_f32_16x16x4_f32(false, a, false, b1, (short)0, acc1, false, false);
        acc2 = __builtin_amdgcn_wmma_f32_16x16x4_f32(false, a, false, b2, (short)0, acc2, false, false);
        acc3 = __builtin_amdgcn_wmma_f32_16x16x4_f32(false, a, false, b3, (short)0, acc3, false, false);
    }

    v8f* accs[4] = { &acc0, &acc1, &acc2, &acc3 };
#pragma unroll
    for (int j = 0; j < 4; ++j) {
        const int col = n0 + j * 16 + l16;
        const float bval = bias ? bias[col] : 0.0f;
#pragma unroll
        for (int r = 0; r < 8; ++r) {
            const int row = m0 + r + 8 * halfsel;
            const size_t idx = (size_t)row * NDIM + col;
            float val = (*accs[j])[r] + bval;
            if (accum) val += Out[idx];
            Out[idx] = val;
        }
    }
}

// ---------------------------------------------------------------------------
// Attention core: for each (b,t) block (4096 blocks, 128 threads):
//   scores[s] = sum_d v[d]*tanh(wq[b,t,d] + uh[b,s,d])   (lane-per-s, d-loop,
//               v_tanh_f32 on TRANS pipe co-executing with VALU FMAs)
//   align    = softmax_s(scores)                          (v_exp_f32)
//   c[b,t,d] = sum_s align[s]*mem[b,s,d]                 (thread-per-4d, s-loop)
// wq row and v broadcast from LDS; uh rows streamed as float4 (L2-resident).
// ---------------------------------------------------------------------------
__global__ __launch_bounds__(128)
void sattn_attn_kernel(const float* __restrict__ wq,
                       const float* __restrict__ uh,
                       const float* __restrict__ vvec,
                       const float* __restrict__ mem,
                       float* __restrict__ cbuf)
{
    __shared__ float wq_s[KDIM];
    __shared__ float v_s[KDIM];
    __shared__ float align_s[LSEQ];
    __shared__ float red[LSEQ];

    const int t    = threadIdx.x;        // 0..127
    const int b    = blockIdx.x >> 7;    // / 128
    const int trow = blockIdx.x & 127;

    const size_t rowbase = ((size_t)b * LSEQ + trow) * KDIM;

    // stage wq row and v into LDS (float4 per thread)
    *(float4*)(wq_s + t * 4) = *(const float4*)(wq + rowbase + t * 4);
    *(float4*)(v_s + t * 4)  = *(const float4*)(vvec + t * 4);
    __syncthreads();

    // ---- scores: lane handles s = t ----
    const float* uhrow = uh + ((size_t)b * LSEQ + t) * KDIM;
    float score = 0.0f;
    for (int d = 0; d < KDIM; d += 4) {
        float4 u = *(const float4*)(uhrow + d);
        float4 x;
        x.x = wq_s[d + 0] + u.x;
        x.y = wq_s[d + 1] + u.y;
        x.z = wq_s[d + 2] + u.z;
        x.w = wq_s[d + 3] + u.w;
        float4 th = htanh4(x);
        score += v_s[d + 0] * th.x;
        score += v_s[d + 1] * th.y;
        score += v_s[d + 2] * th.z;
        score += v_s[d + 3] * th.w;
    }

    // ---- softmax over 128 ----
    red[t] = score;
    __syncthreads();
    for (int off = 64; off > 0; off >>= 1) {
        if (t < off) red[t] = fmaxf(red[t], red[t + off]);
        __syncthreads();
    }
    const float mx = red[0];
    __syncthreads();
    const float e = __expf(score - mx);
    red[t] = e;
    __syncthreads();
    for (int off = 64; off > 0; off >>= 1) {
        if (t < off) red[t] = red[t] + red[t + off];
        __syncthreads();
    }
    align_s[t] = e / red[0];
    __syncthreads();

    // ---- context: thread t handles d = 4t..4t+3 (coalesced float4) ----
    float4 acc = make_float4(0.f, 0.f, 0.f, 0.f);
    const float* memb = mem + (size_t)b * LSEQ * KDIM + t * 4;
    for (int s = 0; s < LSEQ; ++s) {
        const float a = align_s[s];
        float4 mv = *(const float4*)(memb + (size_t)s * KDIM);
        acc.x += a * mv.x; acc.y += a * mv.y;
        acc.z += a * mv.z; acc.w += a * mv.w;
    }
    *(float4*)(cbuf + rowbase + t * 4) = acc;
}

// ---------------------------------------------------------------------------
// Final mix: out = (1-m)*t1 + m*t2   (m read from device scalar)
// ---------------------------------------------------------------------------
__global__ __launch_bounds__(256)
void sattn_mix_kernel(const float* __restrict__ t1,
                      const float* __restrict__ t2,
                      const float* __restrict__ mixp,
                      float* __restrict__ out,
                      int n4)
{
    const int i = blockIdx.x * blockDim.x + threadIdx.x;
    if (i >= n4) return;
    const float m = mixp[0];
    float4 a = ((const float4*)t1)[i];
    float4 b = ((const float4*)t2)[i];
    float4 r;
    r.x = (1.0f - m) * a.x + m * b.x;
    r.y = (1.0f - m) * a.y + m * b.y;
    r.z = (1.0f - m) * a.z + m * b.z;
    r.w = (1.0f - m) * a.w + m * b.w;
    ((float4*)out)[i] = r;
}

extern "C" void kernel_launch(void* const* d_in, const int* in_sizes, int n_in,
                              void* d_out, int out_size, void* d_ws, size_t ws_size,
                              hipStream_t stream) {
    const float* sotw = (const float*)d_in[0];  // current_sotw [B,L,D]
    const float* nobs = (const float*)d_in[1];  // new_obs      [B,L,D]
    const float* Wq   = (const float*)d_in[2];  // [D,D]
    const float* bq   = (const float*)d_in[3];  // [D]
    const float* Wc   = (const float*)d_in[4];  // [D,D]
    const float* vv   = (const float*)d_in[5];  // [D]
    const float* Wout = (const float*)d_in[6];  // [2D,D]
    const float* bout = (const float*)d_in[7];  // [D]
    const float* mix  = (const float*)d_in[8];  // [1]
    float* out = (float*)d_out;

    const float* WoutTop = Wout;                        // rows 0..D-1   (x c)
    const float* WoutBot = Wout + (size_t)KDIM * NDIM;  // rows D..2D-1  (x source)

    // workspace partition: 5 buffers of M*N floats (8 MB each)
    const size_t NLD = (size_t)MDIM * NDIM;
    float* ws   = (float*)d_ws;
    float* wqb  = ws;            // wq (reused dir1/dir2)
    float* uhb  = ws + 1 * NLD;  // uh (reused)
    float* cb   = ws + 2 * NLD;  // context (reused)
    float* t1   = ws + 3 * NLD;  // attn_h dir 1
    float* t2   = ws + 4 * NLD;  // attn_h dir 2

    dim3 ggrid(MDIM / 64, NDIM / 64);
    dim3 gblk(128);

    // ---- direction 1: source = new_obs, memory = current_sotw ----
    sattn_gemm_wmma_f32<<<ggrid, gblk, 0, stream>>>(nobs, Wq, bq,      wqb, 0);
    sattn_gemm_wmma_f32<<<ggrid, gblk, 0, stream>>>(sotw, Wc, nullptr, uhb, 0);
    sattn_attn_kernel<<<MDIM, 128, 0, stream>>>(wqb, uhb, vv, sotw, cb);
    sattn_gemm_wmma_f32<<<ggrid, gblk, 0, stream>>>(cb,   WoutTop, bout,    t1, 0);
    sattn_gemm_wmma_f32<<<ggrid, gblk, 0, stream>>>(nobs, WoutBot, nullptr, t1, 1);

    // ---- direction 2: source = current_sotw, memory = new_obs ----
    sattn_gemm_wmma_f32<<<ggrid, gblk, 0, stream>>>(sotw, Wq, bq,      wqb, 0);
    sattn_gemm_wmma_f32<<<ggrid, gblk, 0, stream>>>(nobs, Wc, nullptr, uhb, 0);
    sattn_attn_kernel<<<MDIM, 128, 0, stream>>>(wqb, uhb, vv, nobs, cb);
    sattn_gemm_wmma_f32<<<ggrid, gblk, 0, stream>>>(cb,   WoutTop, bout,    t2, 0);
    sattn_gemm_wmma_f32<<<ggrid, gblk, 0, stream>>>(sotw, WoutBot, nullptr, t2, 1);

    // ---- mix ----
    const int n4 = (int)(NLD / 4);
    sattn_mix_kernel<<<(n4 + 255) / 256, 256, 0, stream>>>(t1, t2, mix, out, n4);
}